// KGAT_1675037246210
// MI455X (gfx1250) — compile-verified
//
#include <hip/hip_runtime.h>
#include <math.h>

typedef __attribute__((ext_vector_type(2))) float v2f;
typedef __attribute__((ext_vector_type(8))) float v8f;

#define N_NODES_C 100000
#define EMBED_C 64
#define OUT_STRIDE 112
#define EPS_C 1e-12f

__device__ __forceinline__ float lrelu(float x) {
    return x > 0.0f ? x : 0.01f * x;
}

// ---------------------------------------------------------------- zero
__global__ void zero_f32(float* __restrict__ p, int n) {
    int i = blockIdx.x * blockDim.x + threadIdx.x;
    if (i < n) p[i] = 0.0f;
}

// ---------------------------------------------------------------- fuse: ego = emb * (1 + tanh(aux @ auxW^T + auxb))
__global__ void fuse_kernel(const float* __restrict__ aux, const float* __restrict__ emb,
                            const float* __restrict__ auxW, const float* __restrict__ auxb,
                            float* __restrict__ ego, float* __restrict__ out) {
    int gid = blockIdx.x * blockDim.x + threadIdx.x;
    if (gid >= N_NODES_C * EMBED_C) return;
    int n = gid >> 6;
    int j = gid & 63;
    float a0 = aux[n * 3 + 0], a1 = aux[n * 3 + 1], a2 = aux[n * 3 + 2];
    float d = a0 * auxW[j * 3 + 0] + a1 * auxW[j * 3 + 1] + a2 * auxW[j * 3 + 2] + auxb[j];
    float e = emb[gid] * (1.0f + tanhf(d));
    ego[gid] = e;
    out[(size_t)n * OUT_STRIDE + j] = e;
}

// ---------------------------------------------------------------- sparse aggregation: side[row] += val * ego[col]
template <int D>
__global__ void edge_kernel(const float* __restrict__ ego, const float* __restrict__ vals,
                            const int* __restrict__ rows, const int* __restrict__ cols,
                            float* __restrict__ side, int n_edges) {
    unsigned long long tid = (unsigned long long)blockIdx.x * blockDim.x + threadIdx.x;
    unsigned long long total = (unsigned long long)n_edges * D;
    if (tid >= total) return;
    unsigned int e = (unsigned int)(tid / (unsigned)D);
    unsigned int c = (unsigned int)(tid % (unsigned)D);
    int r = rows[e];
    int cl = cols[e];
    float v = vals[e];
    // L2-resident fp32 hardware atomic (ego/side tiles fit in MI455X 192MB L2)
    unsafeAtomicAdd(&side[(size_t)r * D + c], v * ego[(size_t)cl * D + c]);
}

// ---------------------------------------------------------------- bi-interaction layer with fp32 WMMA
// ego_next = lrelu((ego+side)@W1^T + b1) + lrelu((ego*side)@W2^T + b2)
// out[:, col_base:col_base+DOUT] = l2norm(ego_next)
template <int DIN, int DOUT>
__global__ __launch_bounds__(32 * (DOUT / 16)) void layer_kernel(
    const float* __restrict__ ego, const float* __restrict__ side,
    const float* __restrict__ W1, const float* __restrict__ b1,
    const float* __restrict__ W2, const float* __restrict__ b2,
    float* __restrict__ ego_next,  // may be nullptr (last layer)
    float* __restrict__ out, int out_col_base) {
    constexpr int NW = DOUT / 16;     // waves per block: one 16-col output tile each
    constexpr int TPB = 32 * NW;
    __shared__ float h1[16][DIN + 1]; // +1 pad: stride 65/33 banks -> conflict-free column reads
    __shared__ float h2[16][DIN + 1];
    __shared__ float ot[16][DOUT];
    __shared__ float nrm[16];

    const int tid = threadIdx.x;
    const int base = blockIdx.x * 16;  // 100000 = 6250 * 16, exact

    for (int idx = tid; idx < 16 * DIN; idx += TPB) {
        int r = idx / DIN, c = idx % DIN;
        float e = ego[(size_t)(base + r) * DIN + c];
        float s = side[(size_t)(base + r) * DIN + c];
        h1[r][c] = e + s;  // sum branch input
        h2[r][c] = e * s;  // bi branch input
    }
    __syncthreads();

    const int lane = tid & 31;
    const int wave = tid >> 5;
    const int n0 = wave * 16;              // this wave's output-column tile
    const int mr = lane & 15;              // A row / B column within tile
    const int kh = (lane < 16) ? 0 : 2;    // K half per ISA fp32 A/B layout
    const int ncol = n0 + mr;              // global output column for B fragment

    v8f c1 = {};
    v8f c2 = {};
    // K loop: DIN/4 fp32 WMMA steps per branch (uniform control flow, EXEC all ones)
    for (int k = 0; k < DIN / 4; ++k) {
        int kb = k * 4 + kh;
        v2f a1, a2, bw1, bw2;
        a1.x = h1[mr][kb];               a1.y = h1[mr][kb + 1];
        a2.x = h2[mr][kb];               a2.y = h2[mr][kb + 1];
        bw1.x = W1[ncol * DIN + kb];     bw1.y = W1[ncol * DIN + kb + 1];  // B[k][n] = W1[n][k]
        bw2.x = W2[ncol * DIN + kb];     bw2.y = W2[ncol * DIN + kb + 1];
        c1 = __builtin_amdgcn_wmma_f32_16x16x4_f32(false, a1, false, bw1, (short)0, c1, false, false);
        c2 = __builtin_amdgcn_wmma_f32_16x16x4_f32(false, a2, false, bw2, (short)0, c2, false, false);
    }

    const float bias1 = b1[ncol];
    const float bias2 = b2[ncol];
    const int mbase = (lane < 16) ? 0 : 8;  // C/D layout: VGPR r -> M = r (+8 for high lanes)
#pragma unroll
    for (int r = 0; r < 8; ++r) {
        float v = lrelu(c1[r] + bias1) + lrelu(c2[r] + bias2);
        ot[mbase + r][ncol] = v;
    }
    __syncthreads();

    if (tid < 16) {
        float s = 0.0f;
        for (int c = 0; c < DOUT; ++c) {
            float v = ot[tid][c];
            s += v * v;
        }
        nrm[tid] = fmaxf(sqrtf(s), EPS_C);
    }
    __syncthreads();

    for (int idx = tid; idx < 16 * DOUT; idx += TPB) {
        int r = idx / DOUT, c = idx % DOUT;
        float v = ot[r][c];
        if (ego_next) ego_next[(size_t)(base + r) * DOUT + c] = v;  // raw feeds next layer
        out[(size_t)(base + r) * OUT_STRIDE + out_col_base + c] = v / nrm[r];
    }
}

// ---------------------------------------------------------------- launch
extern "C" void kernel_launch(void* const* d_in, const int* in_sizes, int n_in,
                              void* d_out, int out_size, void* d_ws, size_t ws_size,
                              hipStream_t stream) {
    const float* aux  = (const float*)d_in[0];
    const float* emb  = (const float*)d_in[1];
    const float* auxW = (const float*)d_in[2];
    const float* auxb = (const float*)d_in[3];
    const float* evals = (const float*)d_in[4];
    const int*   erows = (const int*)d_in[5];
    const int*   ecols = (const int*)d_in[6];
    const float* W1_0 = (const float*)d_in[7];
    const float* b1_0 = (const float*)d_in[8];
    const float* W2_0 = (const float*)d_in[9];
    const float* b2_0 = (const float*)d_in[10];
    const float* W1_1 = (const float*)d_in[11];
    const float* b1_1 = (const float*)d_in[12];
    const float* W2_1 = (const float*)d_in[13];
    const float* b2_1 = (const float*)d_in[14];
    float* out = (float*)d_out;
    const int n_edges = in_sizes[4];

    // workspace layout (all fp32, ~77 MB total; L2-resident on MI455X)
    float* ego0  = (float*)d_ws;                         // [N,64]
    float* side0 = ego0 + (size_t)N_NODES_C * 64;        // [N,64]
    float* ego1  = side0 + (size_t)N_NODES_C * 64;       // [N,32]
    float* side1 = ego1 + (size_t)N_NODES_C * 32;        // [N,32]

    {
        int n = N_NODES_C * 64;
        zero_f32<<<(n + 255) / 256, 256, 0, stream>>>(side0, n);
        n = N_NODES_C * 32;
        zero_f32<<<(n + 255) / 256, 256, 0, stream>>>(side1, n);
    }
    {
        int n = N_NODES_C * EMBED_C;
        fuse_kernel<<<(n + 255) / 256, 256, 0, stream>>>(aux, emb, auxW, auxb, ego0, out);
    }
    {
        unsigned long long total = (unsigned long long)n_edges * 64ull;
        edge_kernel<64><<<(unsigned)((total + 255) / 256), 256, 0, stream>>>(
            ego0, evals, erows, ecols, side0, n_edges);
    }
    layer_kernel<64, 32><<<N_NODES_C / 16, 64, 0, stream>>>(
        ego0, side0, W1_0, b1_0, W2_0, b2_0, ego1, out, 64);
    {
        unsigned long long total = (unsigned long long)n_edges * 32ull;
        edge_kernel<32><<<(unsigned)((total + 255) / 256), 256, 0, stream>>>(
            ego1, evals, erows, ecols, side1, n_edges);
    }
    layer_kernel<32, 16><<<N_NODES_C / 16, 32, 0, stream>>>(
        ego1, side1, W1_1, b1_1, W2_1, b2_1, nullptr, out, 96);
}